// GCN_66597762891973
// MI455X (gfx1250) — compile-verified
//
#include <hip/hip_runtime.h>
#include <hip/hip_bf16.h>

// ---------------------------------------------------------------------------
// GCN (2-layer, PyG GCNConv) for MI455X / gfx1250, wave32.
//  - Dense GEMMs: V_WMMA_F32_16X16X4_F32 (native fp32 WMMA), W staged in LDS
//    as packed K-pairs so each B fragment pair is one ds_load_2addr_b64.
//  - Aggregate init (h*dinv^2 + bias) fused into the GEMM epilogue.
//  - ReLU is a compile-time template parameter (no per-element cndmask).
//  - Edge scatter: wave-per-edge float4 gathers + global f32 atomics
//    (features + accumulator are L2-resident: 2 x 25.6 MB << 192 MB L2).
// ---------------------------------------------------------------------------

typedef __attribute__((ext_vector_type(2))) float v2f;
typedef __attribute__((ext_vector_type(8))) float v8f;

#define HID 128

// ---------------- degree / normalization ----------------

__global__ void gcn_init_deg(float* __restrict__ deg, int n_nodes) {
    int i = blockIdx.x * blockDim.x + threadIdx.x;
    if (i < n_nodes) deg[i] = 1.0f;
}

__global__ void gcn_deg_accum(const long long* __restrict__ dst, float* __restrict__ deg, int n_edges) {
    int e = blockIdx.x * blockDim.x + threadIdx.x;
    if (e < n_edges) atomicAdd(&deg[(int)dst[e]], 1.0f);
}

__global__ void gcn_dinv(const float* __restrict__ deg, float* __restrict__ dinv, int n_nodes) {
    int i = blockIdx.x * blockDim.x + threadIdx.x;
    if (i < n_nodes) dinv[i] = rsqrtf(deg[i]);
}

// ---------------- fused GEMM + aggregate init -------------------------------
// H[n_rows x 128]   = act(X) @ W             (stored for the edge scatter)
// AGG[n_rows x 128] = H * dinv[row]^2 + bias (self-loop term + bias; the
//                     scatter then atomically accumulates neighbor messages)
//
// One wave per 16-row strip; 8 x (16x16) f32 accumulator tiles cover all 128
// output columns; K consumed in steps of 4 via V_WMMA_F32_16X16X4_F32.
//
// f32 WMMA VGPR layouts (ISA 7.12.2):
//   A (16x4, MxK):  lanes 0-15: {K=k0,K=k0+1} row M=lane; lanes 16-31: {K=k0+2,K=k0+3}
//   B (4x16, KxN):  lanes 0-15: {K=k0,K=k0+1} col N=lane; lanes 16-31: {K=k0+2,K=k0+3}
//   C/D (16x16):    VGPR r: lanes 0-15 -> M=r, N=lane; lanes 16-31 -> M=r+8
//
// W is staged once per block into LDS packed as K-pairs:
//   ldsW[kk*128 + n] = { W[2kk][n], W[2kk+1][n] }   (kk = 0..63)

template <bool RELU>
__global__ void gcn_gemm_fused(const float* __restrict__ X, const float* __restrict__ W,
                               const float* __restrict__ bias, const float* __restrict__ dinv,
                               float* __restrict__ H, float* __restrict__ AGG,
                               int n_rows) {
    __shared__ float2 ldsW[64 * HID];   // 64 KB

    // cooperative W preload + K-pair packing (all threads, before any exit)
    for (int idx = threadIdx.x; idx < 64 * HID; idx += blockDim.x) {
        const int kk = idx >> 7;
        const int n  = idx & (HID - 1);
        float2 p;
        p.x = W[(size_t)(2 * kk)     * HID + n];
        p.y = W[(size_t)(2 * kk + 1) * HID + n];
        ldsW[idx] = p;
    }
    __syncthreads();

    const int wave    = (int)((blockIdx.x * blockDim.x + threadIdx.x) >> 5);
    const int lane    = threadIdx.x & 31;
    const int n_tiles = (n_rows + 15) >> 4;
    if (wave >= n_tiles) return;          // wave-uniform: EXEC stays all-1s for WMMA

    const int m0   = wave << 4;
    const int half = lane >> 4;           // 0: K pair {0,1}; 1: K pair {2,3}
    const int l15  = lane & 15;
    int arow = m0 + l15;
    if (arow > n_rows - 1) arow = n_rows - 1;   // defensive (n_rows % 16 == 0 here)

    v8f acc[8];
    #pragma unroll
    for (int t = 0; t < 8; ++t)
        #pragma unroll
        for (int r = 0; r < 8; ++r) acc[t][r] = 0.0f;

    const float* __restrict__ xrow = X + (size_t)arow * HID;

    for (int k0 = 0; k0 < HID; k0 += 4) {
        const int ka = k0 + (half << 1);
        v2f a;
        a.x = xrow[ka];
        a.y = xrow[ka + 1];
        if (RELU) {
            a.x = fmaxf(a.x, 0.0f);
            a.y = fmaxf(a.y, 0.0f);
        }
        const float2* __restrict__ brow = &ldsW[(size_t)((k0 >> 1) + half) * HID];
        #pragma unroll
        for (int t = 0; t < 8; ++t) {
            const float2 bp = brow[(t << 4) + l15];
            v2f b;
            b.x = bp.x;
            b.y = bp.y;
            acc[t] = __builtin_amdgcn_wmma_f32_16x16x4_f32(
                false, a, false, b, (short)0, acc[t], false, false);
        }
    }

    // fused epilogue: store H and AGG = H*dinv^2 + bias
    float bcol[8];
    #pragma unroll
    for (int t = 0; t < 8; ++t) bcol[t] = bias[(t << 4) + l15];

    const int mbase = m0 + (half << 3);   // lane half selects rows r / r+8
    #pragma unroll
    for (int r = 0; r < 8; ++r) {
        const int row = mbase + r;
        const float di = dinv[row];
        const float dd = di * di;
        float* __restrict__ hrow = H   + (size_t)row * HID + l15;
        float* __restrict__ grow = AGG + (size_t)row * HID + l15;
        #pragma unroll
        for (int t = 0; t < 8; ++t) {
            const float v = acc[t][r];
            hrow[t << 4] = v;
            grow[t << 4] = v * dd + bcol[t];
        }
    }
}

// ---------------- edge scatter: out[dst] += h[src] * dinv[src]*dinv[dst] ----
// One wave32 per edge; each lane moves 4 floats (float4 gather + 4 f32 atomics).

__global__ void gcn_edge_scatter(const long long* __restrict__ src,
                                 const long long* __restrict__ dst,
                                 const float* __restrict__ h,
                                 const float* __restrict__ dinv,
                                 float* __restrict__ out, int n_edges) {
    const int e    = (int)((blockIdx.x * blockDim.x + threadIdx.x) >> 5);
    const int lane = threadIdx.x & 31;
    if (e >= n_edges) return;
    const int s = (int)src[e];
    const int d = (int)dst[e];
    const float w = dinv[s] * dinv[d];
    const float4 v = ((const float4*)(h + (size_t)s * HID))[lane];
    float* __restrict__ od = out + (size_t)d * HID + (lane << 2);
    atomicAdd(od + 0, v.x * w);
    atomicAdd(od + 1, v.y * w);
    atomicAdd(od + 2, v.z * w);
    atomicAdd(od + 3, v.w * w);
}

// ---------------------------------------------------------------------------

extern "C" void kernel_launch(void* const* d_in, const int* in_sizes, int n_in,
                              void* d_out, int out_size, void* d_ws, size_t ws_size,
                              hipStream_t stream) {
    const float*     x  = (const float*)d_in[0];
    const long long* ei = (const long long*)d_in[1];
    const float*     W1 = (const float*)d_in[2];
    const float*     b1 = (const float*)d_in[3];
    const float*     W2 = (const float*)d_in[4];
    const float*     b2 = (const float*)d_in[5];

    const int n_nodes = in_sizes[0] / HID;
    const int n_edges = in_sizes[1] / 2;
    const long long* src = ei;
    const long long* dst = ei + n_edges;

    // workspace layout (floats), node vectors padded to 128-float alignment
    const size_t npad = ((size_t)n_nodes + 127) & ~(size_t)127;
    float* deg  = (float*)d_ws;
    float* dinv = deg + npad;
    float* hbuf = dinv + npad;                       // h1, later reused as h2
    float* agg1 = hbuf + (size_t)n_nodes * HID;      // layer-1 aggregate
    float* zout = (float*)d_out;                     // layer-2 aggregate (final)

    const int tpb = 256;
    const int node_blocks = (n_nodes + tpb - 1) / tpb;
    const int edge_blocks = (n_edges + tpb - 1) / tpb;
    const int n_tiles     = (n_nodes + 15) / 16;
    const int gemm_blocks = (n_tiles + (tpb / 32) - 1) / (tpb / 32);   // 8 waves/block
    const int scat_blocks = ((n_edges * 32) + tpb - 1) / tpb;          // wave per edge

    // normalization (shared by both layers)
    gcn_init_deg <<<node_blocks, tpb, 0, stream>>>(deg, n_nodes);
    gcn_deg_accum<<<edge_blocks, tpb, 0, stream>>>(dst, deg, n_edges);
    gcn_dinv     <<<node_blocks, tpb, 0, stream>>>(deg, dinv, n_nodes);

    // ---- layer 1: h1 = x@W1 ; agg1 = h1*dinv^2 + b1 ; scatter into agg1 ----
    gcn_gemm_fused<false><<<gemm_blocks, tpb, 0, stream>>>(x, W1, b1, dinv, hbuf, agg1, n_nodes);
    gcn_edge_scatter     <<<scat_blocks, tpb, 0, stream>>>(src, dst, hbuf, dinv, agg1, n_edges);

    // ---- layer 2: h2 = relu(agg1)@W2 ; zout = h2*dinv^2 + b2 ; scatter ----
    gcn_gemm_fused<true> <<<gemm_blocks, tpb, 0, stream>>>(agg1, W2, b2, dinv, hbuf, zout, n_nodes);
    gcn_edge_scatter     <<<scat_blocks, tpb, 0, stream>>>(src, dst, hbuf, dinv, zout, n_edges);
}